// LIFNode_48455821033991
// MI455X (gfx1250) — compile-verified
//
#include <hip/hip_runtime.h>

// LIF forward scan, X:[B=64, T=32, N=65536] f32 -> spikes, same shape.
//   mem = 2*x_t - mem ; spike = mem>1 ; mem = spike ? 0 : mem
// HBM-bound (1 GiB single-touch, floor ~46us @ 23.3 TB/s). The t-walk is a
// strided 2D tile gather -> fetch it with the Tensor Data Mover: one
// tensor_load_to_lds per 64KiB [T x W] tile, double-buffered in LDS.

typedef float        v2f __attribute__((ext_vector_type(2)));
typedef float        v4f __attribute__((ext_vector_type(4)));
typedef unsigned int v4u __attribute__((ext_vector_type(4)));
typedef int          v4i __attribute__((ext_vector_type(4)));
typedef int          v8i __attribute__((ext_vector_type(8)));

static constexpr int   LIF_B = 64;
static constexpr int   LIF_T = 32;
static constexpr int   LIF_N = 65536;
static constexpr float V_TH  = 1.0f;

static constexpr int W               = 512;                    // tile width (floats)
static constexpr int TILES_PER_B     = LIF_N / W;              // 128
static constexpr int TILES           = LIF_B * TILES_PER_B;    // 8192
static constexpr int TILES_PER_BLOCK = 8;
static constexpr int BLOCKS          = TILES / TILES_PER_BLOCK;// 1024
static constexpr int THREADS         = 256;                    // 8 waves (wave32)
static constexpr int TILE_BYTES      = LIF_T * W * 4;          // 64 KiB
static constexpr int LDS_BYTES       = 2 * TILE_BYTES;         // 128 KiB (dynamic)

#if __has_builtin(__builtin_amdgcn_tensor_load_to_lds) && \
    __has_builtin(__builtin_amdgcn_s_wait_tensorcnt)
#define LIF_USE_TDM 1
#else
#define LIF_USE_TDM 0
#endif

#if LIF_USE_TDM

// Issue one TDM 2D tile load: tile g -> LDS buffer `buf`.
// D# per CDNA5 ISA ch.8: group0 = {count/flags, lds_addr, global_addr, type=2},
// group1 = geometry, groups 2/3 unused (2D tensor).
// This toolchain's builtin is the 6-arg form:
//   (uint32x4 g0, int32x8 g1, int32x4, int32x4, int32x8, i32 cpol)
__device__ __forceinline__ void issue_tile_tdm(unsigned long long xbase, int g, int buf,
                                               v8i g1, v4i g2, v4i g3, v8i g4)
{
    const int b  = g >> 7;          // g / TILES_PER_B (128)
    const int nt = g & (TILES_PER_B - 1);
    const unsigned long long gaddr =
        xbase + ((unsigned long long)b * ((unsigned long long)LIF_T * LIF_N)
               + (unsigned long long)nt * W) * 4ull;

    v4u g0;
    g0[0] = 1u;                                              // count=1 (valid), user mode
    g0[1] = (unsigned)buf * (unsigned)TILE_BYTES;            // lds_addr (bytes; dyn LDS @0)
    g0[2] = (unsigned)(gaddr & 0xFFFFFFFFull);               // global_addr[31:0]
    g0[3] = (unsigned)((gaddr >> 32) & 0x01FFFFFFull)        // global_addr[56:32]
          | 0x80000000u;                                     // type=2 ("image")

    __builtin_amdgcn_tensor_load_to_lds(g0, g1, g2, g3, g4, 0);
}

__global__ __launch_bounds__(THREADS) void lif_tdm_kernel(
    const float* __restrict__ X, float* __restrict__ out)
{
    // Dynamic LDS: external addrspace(3) symbol -> contents are opaque to the
    // optimizer, so reads below cannot be folded away even though the TDM
    // builtin carries no pointer operand. Also starts at LDS offset 0,
    // matching the lds_addr fields in the descriptors.
    extern __shared__ float sh[];                            // 2 x 64 KiB buffers

    const int tid = threadIdx.x;
    const unsigned long long xbase = (unsigned long long)(const void*)X;

    // D# group1 (constant geometry): data_size=4B(code 2), tensor_dim0=W,
    // tensor_dim1=T, tile_dim0=W, tile_dim1=T, tile_dim2=0, dim0_stride=N.
    v8i g1;
    g1[0] = (2 << 16);                       // data_size=2 (4 bytes)
    g1[1] = (W & 0xFFFF) << 16;              // tensor_dim0[15:0]  @ bits[63:48]
    g1[2] = (LIF_T & 0xFFFF) << 16;          // tensor_dim1[15:0]  @ bits[95:80]
    g1[3] = (W & 0xFFFF) << 16;              // tile_dim0          @ bits[127:112]
    g1[4] = (LIF_T & 0xFFFF);                // tile_dim1 @ [143:128]; tile_dim2=0
    g1[5] = LIF_N;                           // tensor_dim0_stride[31:0]
    g1[6] = 0;                               // stride0[47:32]=0, stride1[15:0]=0
    g1[7] = 0;
    const v4i g2 = {0, 0, 0, 0};             // 2D: groups 2/3 unused
    const v4i g3 = {0, 0, 0, 0};
    const v8i g4 = {0, 0, 0, 0, 0, 0, 0, 0}; // 5th operand (toolchain-specific): zero

    const int tile0 = blockIdx.x * TILES_PER_BLOCK;

    if (tid < 32) {                          // wave 0 drives the DMA pipeline
        issue_tile_tdm(xbase, tile0, 0, g1, g2, g3, g4);
    }

    for (int i = 0; i < TILES_PER_BLOCK; ++i) {
        if (tid < 32) {
            if (i + 1 < TILES_PER_BLOCK) {
                issue_tile_tdm(xbase, tile0 + i + 1, (i + 1) & 1, g1, g2, g3, g4);
                __builtin_amdgcn_s_wait_tensorcnt(1);  // in-order: tile i landed
            } else {
                __builtin_amdgcn_s_wait_tensorcnt(0);  // last tile landed
            }
        }
        __syncthreads();                     // tile i visible to all 8 waves
        asm volatile("" ::: "memory");       // pin DS reads after the wait/barrier

        // Scan tile i: each thread owns 2 consecutive neurons across all T.
        const int g  = tile0 + i;
        const int b  = g >> 7;
        const int nt = g & (TILES_PER_B - 1);
        const float* shbuf = &sh[(i & 1) * (LIF_T * W)];   // stays addrspace(3)
        float* __restrict__ obase =
            out + ((long)b * LIF_T * LIF_N + (long)nt * W + tid * 2);

        float m0 = 0.0f, m1 = 0.0f;
        #pragma unroll 4
        for (int t = 0; t < LIF_T; ++t) {
            const v2f x = *(const v2f*)&shbuf[t * W + tid * 2];   // ds_load_b64
            m0 = 2.0f * x[0] - m0;                                // tau = 0.5
            m1 = 2.0f * x[1] - m1;
            v2f s;
            const bool f0 = m0 > V_TH, f1 = m1 > V_TH;
            s[0] = f0 ? 1.0f : 0.0f;  m0 = f0 ? 0.0f : m0;        // hard reset
            s[1] = f1 ? 1.0f : 0.0f;  m1 = f1 ? 0.0f : m1;
            __builtin_nontemporal_store(s, (v2f*)(obase + (long)t * LIF_N));
        }
        __syncthreads();                     // reads done before buffer reuse
    }
}

#else  // fallback: direct streaming kernel (round-1 design)

__global__ __launch_bounds__(THREADS) void lif_scan_kernel(
    const float* __restrict__ X, float* __restrict__ out, long total_cols)
{
    const long tid = (long)blockIdx.x * blockDim.x + threadIdx.x;
    if (tid >= total_cols) return;

    const int  cols_per_b = LIF_N / 4;
    const int  b   = (int)(tid / cols_per_b);
    const int  n4  = (int)(tid % cols_per_b);
    const long base = (long)b * LIF_T * LIF_N + (long)n4 * 4;

    const v4f* __restrict__ src = (const v4f*)(X + base);
    v4f*       __restrict__ dst = (v4f*)(out + base);
    const long strideV = LIF_N / 4;

    v4f mem = (v4f){0.f, 0.f, 0.f, 0.f};
    #pragma unroll 4
    for (int t = 0; t < LIF_T; ++t) {
        if (t + 2 < LIF_T)
            __builtin_prefetch((const void*)(src + (long)(t + 2) * strideV), 0, 0);
        v4f x = __builtin_nontemporal_load(src + (long)t * strideV);
        v4f m = x * 2.0f - mem;
        v4f s;
        #pragma unroll
        for (int k = 0; k < 4; ++k) {
            const float mv = m[k];
            const bool  fire = mv > V_TH;
            s[k] = fire ? 1.0f : 0.0f;
            m[k] = fire ? 0.0f : mv;
        }
        mem = m;
        __builtin_nontemporal_store(s, dst + (long)t * strideV);
    }
}

#endif

extern "C" void kernel_launch(void* const* d_in, const int* in_sizes, int n_in,
                              void* d_out, int out_size, void* d_ws, size_t ws_size,
                              hipStream_t stream)
{
    const float* X   = (const float*)d_in[0];
    float*       out = (float*)d_out;

#if LIF_USE_TDM
    lif_tdm_kernel<<<BLOCKS, THREADS, LDS_BYTES, stream>>>(X, out);
#else
    const long total_cols = (long)LIF_B * (LIF_N / 4);
    const int  blocks     = (int)((total_cols + THREADS - 1) / THREADS);
    lif_scan_kernel<<<blocks, THREADS, 0, stream>>>(X, out, total_cols);
#endif
}